// PlainTensorProductBlock2_10144712753187
// MI455X (gfx1250) — compile-verified
//
#include <hip/hip_runtime.h>
#include <hip/hip_bf16.h>

// Problem constants (from reference): N=8192, F=128, P1=P2=2.
#define NN 8192
#define FF 128

typedef float f2_t __attribute__((ext_vector_type(2)));
typedef float f4_t __attribute__((ext_vector_type(4)));

// Async 16B copy global -> LDS, tracked on ASYNCcnt (CDNA5 path).
// GV mode: VDST = LDS byte offset (wave-relative), VADDR = 64-bit global addr.
// INST_OFFSET is added to BOTH the LDS and the global address (ISA 08, §4.4),
// so one base register pair serves a whole region via immediate offsets.
__device__ __forceinline__ void async_cp_b128(unsigned lds_off, const float* g) {
    asm volatile("global_load_async_to_lds_b128 %0, %1, off"
                 :: "v"(lds_off), "v"(g) : "memory");
}
__device__ __forceinline__ void async_cp_b128_o4096(unsigned lds_off, const float* g) {
    asm volatile("global_load_async_to_lds_b128 %0, %1, off offset:4096"
                 :: "v"(lds_off), "v"(g) : "memory");
}

__device__ __forceinline__ unsigned lds_off_of(const void* p) {
    // Generic pointer to a __shared__ object: low 32 bits are the LDS byte address.
    return (unsigned)(uintptr_t)p;
}

__global__ __launch_bounds__(256) void
PlainTensorProductBlock2_kernel(const float* __restrict__ x,   // [N,3,F,2]
                                const float* __restrict__ y,   // [N,3,F,2]
                                const float* __restrict__ z,   // [N,F,4]
                                float* __restrict__ out)       // [N,3,3,F,4]
{
    // Per-block staging: 2 n-samples worth of x/y/z, all contiguous in global.
    __shared__ __attribute__((aligned(16))) float sX[2 * 3 * FF * 2]; // 6144 B
    __shared__ __attribute__((aligned(16))) float sY[2 * 3 * FF * 2]; // 6144 B
    __shared__ __attribute__((aligned(16))) float sZ[2 * FF * 4];     // 4096 B

    const int tid = threadIdx.x;
    const int n0  = blockIdx.x * 2;        // two n per 256-thread block

    const float* gx = x + (size_t)n0 * (3 * FF * 2);  // 1536 contiguous floats
    const float* gy = y + (size_t)n0 * (3 * FF * 2);
    const float* gz = z + (size_t)n0 * (FF * 4);      // 1024 contiguous floats

    // ---- async global -> LDS staging (ASYNCcnt), fully unrolled ----
    const unsigned l0 = (unsigned)tid * 16u;      // this thread's 16B slot
    const unsigned lX = lds_off_of(sX) + l0;
    const unsigned lY = lds_off_of(sY) + l0;
    const unsigned lZ = lds_off_of(sZ) + l0;
    const float* px = gx + tid * 4;
    const float* py = gy + tid * 4;
    const float* pz = gz + tid * 4;

    async_cp_b128(lX, px);                 // X bytes [   0, 4096)
    async_cp_b128(lY, py);                 // Y bytes [   0, 4096)
    async_cp_b128(lZ, pz);                 // Z bytes [   0, 4096)  (whole region)
    if (tid < 128) {                       // tail sweeps: 128 lanes x 16B = 2048 B
        async_cp_b128_o4096(lX, px);       // X bytes [4096, 6144)
        async_cp_b128_o4096(lY, py);       // Y bytes [4096, 6144)
    }

    asm volatile("s_wait_asynccnt 0x0" ::: "memory");
    __syncthreads();

    // ---- compute: one (n,f) per thread ----
    const int nl = tid >> 7;        // 0..1
    const int f  = tid & (FF - 1);  // 0..127

    const f2_t* X2 = (const f2_t*)sX;   // [2][3][128] of float2, bank-conflict free
    const f2_t* Y2 = (const f2_t*)sY;
    const f4_t* Z4 = (const f4_t*)sZ;   // [2][128] of float4

    f2_t X[3], Y[3];
#pragma unroll
    for (int a = 0; a < 3; ++a) {
        X[a] = X2[nl * (3 * FF) + a * FF + f];
        Y[a] = Y2[nl * (3 * FF) + a * FF + f];
    }
    const f4_t zv = Z4[nl * FF + f];

    const int n = n0 + nl;
    // out element index: ((n*9 + (a*3+b))*F + f)*4
    float* ob = out + (size_t)n * 9 * (FF * 4) + (size_t)f * 4;

#pragma unroll
    for (int a = 0; a < 3; ++a) {
#pragma unroll
        for (int b = 0; b < 3; ++b) {
            f4_t o;
            // uv flatten: (u,v) -> u*2+v ; o = {(0,0),(0,1),(1,0),(1,1)}
            o.x = 0.75f * (X[a].x * Y[b].x + X[b].x * Y[a].x);
            o.y = 0.75f * (X[a].x * Y[b].y + X[b].x * Y[a].y);
            o.z = 0.75f * (X[a].y * Y[b].x + X[b].y * Y[a].x);
            o.w = 0.75f * (X[a].y * Y[b].y + X[b].y * Y[a].y);
            if (a == b) {   // 0.75 * (2/3) = 0.5 trace-removal on the diagonal
                o.x -= 0.5f * zv.x;
                o.y -= 0.5f * zv.y;
                o.z -= 0.5f * zv.z;
                o.w -= 0.5f * zv.w;
            }
            // Streaming output (written once, never re-read): non-temporal b128.
            __builtin_nontemporal_store(o, (f4_t*)(ob + (size_t)(a * 3 + b) * (FF * 4)));
        }
    }
}

extern "C" void kernel_launch(void* const* d_in, const int* in_sizes, int n_in,
                              void* d_out, int out_size, void* d_ws, size_t ws_size,
                              hipStream_t stream) {
    const float* x = (const float*)d_in[0];   // [N,3,F,2]
    const float* y = (const float*)d_in[1];   // [N,3,F,2]
    const float* z = (const float*)d_in[2];   // [N,F,4]
    // d_in[3] is the 3x3 identity; the Kronecker delta is hardcoded.
    float* out = (float*)d_out;               // [N,3,3,F,4]

    dim3 grid(NN / 2);   // 4096 blocks, 2 n-samples per block
    dim3 block(256);
    PlainTensorProductBlock2_kernel<<<grid, block, 0, stream>>>(x, y, z, out);
}